// LRML_87969520157215
// MI455X (gfx1250) — compile-verified
//
#include <hip/hip_runtime.h>
#include <math.h>

typedef __attribute__((ext_vector_type(2))) float v2f;
typedef __attribute__((ext_vector_type(8))) float v8f;

#define DIM     128
#define DSTRIDE (DIM + 4)   // padded row stride for transposed K (bank spread)
#define NMEM    20          // M (valid memory slots)
#define MPAD    32          // K columns padded (zeros for m >= 20)
#define BUSERS  16
#define WAVES   2
#define TILE    16          // items per wave

// One block = 2 waves = 32 items. Each wave owns a 16-item tile; the block
// stages K^T [32 x 128, zero-padded rows], memories^T [128 x 20], the 16
// gathered user rows, and its 32 item rows into LDS, then loops over the 16
// users running two f32 WMMA GEMMs per user:
//   logits = (item .* u) @ K      (A-frags hoisted, scaling folded into A)
//   rel    = attn @ mem           (accumulator pre-loaded with u - item)
// Transposed B operands make every WMMA B-fragment a single aligned
// ds_load_b64 into an even register pair (no re-pairing moves).
__global__ __launch_bounds__(WAVES * 32) void lrml_score_kernel(
    const float* __restrict__ uemb,   // [N_USERS][128]
    const float* __restrict__ iemb,   // [N_ITEMS][128]
    const float* __restrict__ kmat,   // [128][20] row-major
    const float* __restrict__ mems,   // [20][128] row-major
    const int*   __restrict__ ids,    // [16]
    float*       __restrict__ out,    // [16][N_ITEMS]
    int n_items)
{
    __shared__ __align__(16) float s_items[WAVES * TILE][DIM]; // 16 KB
    __shared__ __align__(16) float s_Kt[MPAD][DSTRIDE];        // 16.5 KB (K^T)
    __shared__ __align__(16) float s_memT[DIM][NMEM];          // 10 KB (mem^T)
    __shared__ __align__(16) float s_u[BUSERS][DIM];           // 8 KB
    __shared__ float s_att[WAVES][TILE][NMEM];                 // logits -> attn
    __shared__ float s_red[WAVES][TILE][16];                   // distance partials

    const int tid  = threadIdx.x;
    const int w    = tid >> 5;        // wave in block
    const int lane = tid & 31;
    const int c16  = lane & 15;       // column / row-within-tile index
    const int half = lane >> 4;       // 0: lanes 0-15, 1: lanes 16-31
    const int kb   = half * 2;        // k-base inside a 4-wide K chunk
    const int itemBase = blockIdx.x * (WAVES * TILE) + w * TILE;

    // ---- stage K transposed, zero-padded to 32 rows ----
    for (int i = tid; i < MPAD * DIM; i += WAVES * 32) {
        const int m = i >> 7, d = i & (DIM - 1);
        s_Kt[m][d] = (m < NMEM) ? kmat[d * NMEM + m] : 0.0f;
    }
    // ---- stage memories transposed: s_memT[d][m] ----
    for (int i = tid; i < NMEM * DIM; i += WAVES * 32) {
        const int m = i >> 7, d = i & (DIM - 1);
        s_memT[d][m] = mems[m * DIM + d];
    }
    // ---- gather the 16 scored user rows ----
    for (int i = tid; i < BUSERS * DIM; i += WAVES * 32) {
        const int b = i >> 7, d = i & (DIM - 1);
        s_u[b][d] = uemb[(long)ids[b] * DIM + d];
    }
    // ---- stage 32 item rows (float4, rows clamped at the tail) ----
    {
        const int base = blockIdx.x * (WAVES * TILE);
        for (int i = tid; i < (WAVES * TILE) * (DIM / 4); i += WAVES * 32) {
            const int r = i / (DIM / 4);
            const int c = i % (DIM / 4);
            int g = base + r;
            if (g >= n_items) g = n_items - 1;
            ((float4*)&s_items[r][0])[c] = ((const float4*)(iemb + (long)g * DIM))[c];
        }
    }
    __syncthreads();

    // ---- hoist user-invariant item A-fragments into registers ----
    // A layout (16x4 f32): lane holds row c16, k = kb + {0,1} of each chunk.
    v2f aItem[DIM / 4];
    #pragma unroll
    for (int c = 0; c < DIM / 4; ++c) {
        aItem[c] = *(const v2f*)&s_items[w * TILE + c16][4 * c + kb];
    }

    #pragma unroll 1
    for (int b = 0; b < BUSERS; ++b) {
        // ======== GEMM1: logits[16 x 20] via V_WMMA_F32_16X16X4_F32 ========
        v8f acc0 = {0.f, 0.f, 0.f, 0.f, 0.f, 0.f, 0.f, 0.f};
        v8f acc1 = {0.f, 0.f, 0.f, 0.f, 0.f, 0.f, 0.f, 0.f};
        #pragma unroll
        for (int c = 0; c < DIM / 4; ++c) {
            const int d0 = 4 * c + kb;
            const v2f uf = *(const v2f*)&s_u[b][d0];
            const v2f a  = aItem[c] * uf;                    // v_pk_mul_f32
            const v2f b0 = *(const v2f*)&s_Kt[c16][d0];      // ds_load_b64
            const v2f b1 = *(const v2f*)&s_Kt[16 + c16][d0]; // ds_load_b64
            acc0 = __builtin_amdgcn_wmma_f32_16x16x4_f32(
                false, a, false, b0, (short)0, acc0, false, false);
            acc1 = __builtin_amdgcn_wmma_f32_16x16x4_f32(
                false, a, false, b1, (short)0, acc1, false, false);
        }
        // scatter logits (C/D layout: lane l, vgpr r -> row r+8*half, col c16)
        #pragma unroll
        for (int r = 0; r < 8; ++r) {
            const int n = r + 8 * half;
            s_att[w][n][c16] = acc0[r];
            if (c16 < NMEM - 16) s_att[w][n][16 + c16] = acc1[r];
        }
        __syncthreads();

        // ======== softmax over m (lane j owns item row j) ========
        if (lane < TILE) {
            float mx = s_att[w][lane][0];
            #pragma unroll
            for (int m = 1; m < NMEM; ++m) mx = fmaxf(mx, s_att[w][lane][m]);
            float e[NMEM];
            float sum = 0.f;
            #pragma unroll
            for (int m = 0; m < NMEM; ++m) {
                e[m] = __expf(s_att[w][lane][m] - mx);
                sum += e[m];
            }
            const float inv = 1.0f / sum;
            #pragma unroll
            for (int m = 0; m < NMEM; ++m) s_att[w][lane][m] = e[m] * inv;
        }
        __syncthreads();

        // ======== GEMM2: (u - item) + attn @ mem, all in the accumulator ===
        // C/D layout: lane l, vgpr r -> row n = r+8*half, col d = 16t + c16.
        v8f rel[8];
        #pragma unroll
        for (int t = 0; t < 8; ++t) {
            const int d = 16 * t + c16;
            const float um = s_u[b][d];
            #pragma unroll
            for (int r = 0; r < 8; ++r) {
                rel[t][r] = um - s_items[w * TILE + r + 8 * half][d];
            }
        }
        #pragma unroll
        for (int mc = 0; mc < NMEM / 4; ++mc) {   // 20 = 5 chunks of K=4
            const int m0 = 4 * mc + kb;
            const v2f a = *(const v2f*)&s_att[w][c16][m0];   // ds_load_b64
            #pragma unroll
            for (int t = 0; t < 8; ++t) {         // 8 d-column tiles of 16
                const v2f bb = *(const v2f*)&s_memT[16 * t + c16][m0];
                rel[t] = __builtin_amdgcn_wmma_f32_16x16x4_f32(
                    false, a, false, bb, (short)0, rel[t], false, false);
            }
        }

        // ======== distance: sum over d of rel^2, reduce across lanes ========
        float pp[8] = {0.f, 0.f, 0.f, 0.f, 0.f, 0.f, 0.f, 0.f};
        #pragma unroll
        for (int t = 0; t < 8; ++t) {
            #pragma unroll
            for (int r = 0; r < 8; ++r) {
                pp[r] += rel[t][r] * rel[t][r];
            }
        }
        #pragma unroll
        for (int r = 0; r < 8; ++r) s_red[w][r + 8 * half][c16] = pp[r];
        __syncthreads();
        if (lane < TILE) {
            float s = 0.f;
            #pragma unroll
            for (int c = 0; c < 16; ++c) s += s_red[w][lane][c];
            const int ng = itemBase + lane;
            if (ng < n_items) out[(long)b * n_items + ng] = -sqrtf(s);
        }
        __syncthreads();
    }
}

extern "C" void kernel_launch(void* const* d_in, const int* in_sizes, int n_in,
                              void* d_out, int out_size, void* d_ws, size_t ws_size,
                              hipStream_t stream) {
    const float* uemb = (const float*)d_in[0];  // [10000][128]
    const float* iemb = (const float*)d_in[1];  // [50000][128]
    const float* kmat = (const float*)d_in[2];  // [128][20]
    const float* mems = (const float*)d_in[3];  // [20][128]
    const int*   ids  = (const int*)d_in[4];    // [16]
    float*       out  = (float*)d_out;          // [16][50000]

    const int n_items = in_sizes[1] / DIM;
    const int items_per_block = WAVES * TILE;   // 32
    const int grid = (n_items + items_per_block - 1) / items_per_block;

    lrml_score_kernel<<<grid, WAVES * 32, 0, stream>>>(
        uemb, iemb, kmat, mems, ids, out, n_items);
}